// MLP_Param_Var_33741263078242
// MI455X (gfx1250) — compile-verified
//
#include <hip/hip_runtime.h>

#define NODE_NUMS 128
#define IN_SIZE   256
#define OUT_SIZE  256
#define BATCH     4096

typedef float v2f __attribute__((ext_vector_type(2)));
typedef float v8f __attribute__((ext_vector_type(8)));

__device__ __forceinline__ int imin(int a, int b) { return a < b ? a : b; }

// ---------------- binning: order[] groups sample ids by expert ----------------

__global__ void k_init(int* __restrict__ cnts, int* __restrict__ curs) {
    int t = threadIdx.x;
    if (t < NODE_NUMS) { cnts[t] = 0; curs[t] = 0; }
}

__global__ void k_count(const int* __restrict__ idx, int* __restrict__ cnts) {
    int b = blockIdx.x * blockDim.x + threadIdx.x;
    if (b < BATCH) atomicAdd(&cnts[idx[b]], 1);
}

__global__ void k_scan(const int* __restrict__ cnts, int* __restrict__ offs) {
    if (threadIdx.x == 0 && blockIdx.x == 0) {
        int acc = 0;
        for (int e = 0; e < NODE_NUMS; ++e) { offs[e] = acc; acc += cnts[e]; }
    }
}

__global__ void k_scatter(const int* __restrict__ idx, const int* __restrict__ offs,
                          int* __restrict__ curs, int* __restrict__ order) {
    int b = blockIdx.x * blockDim.x + threadIdx.x;
    if (b < BATCH) {
        int e = idx[b];
        int p = atomicAdd(&curs[e], 1);
        order[offs[e] + p] = b;
    }
}

// ---------------- grouped GEMM: one expert per block.x, 64 cols per block.y ----
// Wave w handles cols [blockIdx.y*64 + w*16, +16). fp32 WMMA 16x16x4.
// A 16x4 frag (lane L, m=L&15, hi=L>>4): vgpr0 = A[m][k0+2hi], vgpr1 = A[m][k0+2hi+1]
// B 4x16 frag:                           vgpr0 = B[k0+2hi][n], vgpr1 = B[k0+2hi+1][n]
// C/D 16x16: vgpr r = C[r + 8*hi][n]

__global__ __launch_bounds__(128)
void moe_gemm(const float* __restrict__ x, const float* __restrict__ W,
              const float* __restrict__ bias, const int* __restrict__ offs,
              const int* __restrict__ cnts, const int* __restrict__ order,
              float* __restrict__ out) {
    const int e = blockIdx.x;
    const int cnt = cnts[e];
    if (cnt == 0) return;
    const int off  = offs[e];
    const int lane = threadIdx.x & 31;
    const int wave = threadIdx.x >> 5;
    const int nl   = lane & 15;
    const int hi   = lane >> 4;
    const int n0   = blockIdx.y * 64 + wave * 16;

    const float* __restrict__ We = W + (size_t)e * (IN_SIZE * OUT_SIZE);
    const float* __restrict__ Wn = We + (n0 + nl);          // column view, stride OUT_SIZE
    const float  bn = bias[e * OUT_SIZE + n0 + nl];

    for (int mc = 0; mc < cnt; mc += 32) {
        // gather row ids for this lane's A rows (two 16-row tiles); clamp for tails
        const int rid0 = order[off + imin(mc + nl,      cnt - 1)];
        const int rid1 = order[off + imin(mc + 16 + nl, cnt - 1)];
        const float* __restrict__ xr0 = x + (size_t)rid0 * IN_SIZE;
        const float* __restrict__ xr1 = x + (size_t)rid1 * IN_SIZE;

        v8f c0 = {};
        v8f c1 = {};
        for (int k0 = 0; k0 < IN_SIZE; k0 += 4) {
            const int kb = k0 + 2 * hi;
            if (k0 + 40 <= IN_SIZE)   // in-bounds prefetch of the B stream (global_prefetch_b8)
                __builtin_prefetch(Wn + (size_t)(kb + 32) * OUT_SIZE, 0, 0);

            v2f bf;
            bf.x = Wn[(size_t)kb * OUT_SIZE];
            bf.y = Wn[(size_t)(kb + 1) * OUT_SIZE];
            v2f a0 = *(const v2f*)(xr0 + kb);
            v2f a1 = *(const v2f*)(xr1 + kb);

            // D = A*B + C, fp32 end to end (v_wmma_f32_16x16x4_f32); B reused for 2 tiles
            c0 = __builtin_amdgcn_wmma_f32_16x16x4_f32(false, a0, false, bf,
                                                       (short)0, c0, false, false);
            c1 = __builtin_amdgcn_wmma_f32_16x16x4_f32(false, a1, false, bf,
                                                       (short)0, c1, false, false);
        }

        // store valid rows only (EXEC reconverges before next WMMA loop)
        #pragma unroll
        for (int r = 0; r < 8; ++r) {
            const int m0 = mc + r + 8 * hi;
            if (m0 < cnt)
                out[(size_t)order[off + m0] * OUT_SIZE + n0 + nl] = c0[r] + bn;
            const int m1 = mc + 16 + r + 8 * hi;
            if (m1 < cnt)
                out[(size_t)order[off + m1] * OUT_SIZE + n0 + nl] = c1[r] + bn;
        }
    }
}

extern "C" void kernel_launch(void* const* d_in, const int* in_sizes, int n_in,
                              void* d_out, int out_size, void* d_ws, size_t ws_size,
                              hipStream_t stream) {
    const float* x    = (const float*)d_in[0];
    // d_in[1] = var_vector, unused in forward
    const float* W1   = (const float*)d_in[2];
    const float* b1   = (const float*)d_in[3];
    const int*   node = (const int*)d_in[4];
    const int*   idx  = node + BATCH;       // node_ind[1]
    float* out = (float*)d_out;

    int* ws    = (int*)d_ws;
    int* cnts  = ws;            // [128]
    int* offs  = ws + 128;      // [128]
    int* curs  = ws + 256;      // [128]
    int* order = ws + 384;      // [4096]

    k_init   <<<1, 128, 0, stream>>>(cnts, curs);
    k_count  <<<BATCH / 256, 256, 0, stream>>>(idx, cnts);
    k_scan   <<<1, 32, 0, stream>>>(cnts, offs);
    k_scatter<<<BATCH / 256, 256, 0, stream>>>(idx, offs, curs, order);

    dim3 grid(NODE_NUMS, OUT_SIZE / 64);
    moe_gemm <<<grid, 128, 0, stream>>>(x, W1, b1, offs, cnts, order, out);
}